// NAISAttention_55284819034150
// MI455X (gfx1250) — compile-verified
//
#include <hip/hip_runtime.h>
#include <hip/hip_bf16.h>

typedef __attribute__((ext_vector_type(2))) float v2f;
typedef __attribute__((ext_vector_type(8))) float v8f;

// Problem constants (from reference)
#define BATCH   1024
#define LSEQ    200
#define LPAD    208        // 13 * 16
#define DDIM    256
#define MTILES  13
#define ROWP    260        // 256 + 4-float pad: conflict-free 16-row fragment reads
#define WCHUNK  64         // W rows staged per LDS chunk (4 chunks of 64 = 256)

// LDS layout (float offsets)
#define CTX_OFF   0
#define CTX_SZ    (LPAD * ROWP)            // 54080
#define WBUF_OFF  (CTX_OFF + CTX_SZ)       // 54080
#define WBUF_SZ   (WCHUNK * ROWP)          // 16640
#define QV_OFF    (WBUF_OFF + WBUF_SZ)     // 70720
#define MK_OFF    (QV_OFF + DDIM)          // 70976
#define BV_OFF    (MK_OFF + LPAD)          // 71184
#define HV_OFF    (BV_OFF + DDIM)          // 71440
#define AB_OFF    (HV_OFF + DDIM)          // 71696
#define AT_OFF    (AB_OFF + LPAD)          // 71904
#define RED_OFF   (AT_OFF + LPAD)          // 72112
#define SMEM_FLOATS (RED_OFF + 16)         // 72128
#define SMEM_BYTES  (SMEM_FLOATS * 4)      // 288512 B  (< 320 KB WGP LDS)

__global__ __launch_bounds__(256)
void nais_attention_fused(const float* __restrict__ query,    // [B,D]
                          const float* __restrict__ context,  // [B,L,D]
                          const float* __restrict__ cmask,    // [B,L]
                          const float* __restrict__ W,        // [D,D] row-major [e][d]
                          const float* __restrict__ bias,     // [D]
                          const float* __restrict__ hvec,     // [D,1]
                          float* __restrict__ out)            // [B*D] output | [B*L] attn
{
    extern __shared__ __align__(16) float smem[];
    float* ctx   = smem + CTX_OFF;   // masked context, row stride ROWP
    float* wbuf  = smem + WBUF_OFF;  // q-scaled W chunk, row stride ROWP
    float* qv    = smem + QV_OFF;
    float* mk    = smem + MK_OFF;
    float* bv    = smem + BV_OFF;
    float* hv    = smem + HV_OFF;
    float* Abuf  = smem + AB_OFF;
    float* attns = smem + AT_OFF;
    float* red   = smem + RED_OFF;

    const int b    = blockIdx.x;
    const int tid  = threadIdx.x;
    const int lane = tid & 31;
    const int wave = tid >> 5;

    // ---- stage small vectors ----
    qv[tid] = query[(size_t)b * DDIM + tid];
    bv[tid] = bias[tid];
    hv[tid] = hvec[tid];
    if (tid < LPAD) {
        mk[tid]   = (tid < LSEQ) ? cmask[(size_t)b * LSEQ + tid] : 0.0f;
        Abuf[tid] = 0.0f;
    }
    __syncthreads();

    // ---- stage masked context: 200*256 floats, float4 global loads ----
    {
        const float4* src = (const float4*)(context + (size_t)b * LSEQ * DDIM);
        for (int i = tid; i < (LSEQ * DDIM / 4); i += 256) {
            float4 v = src[i];
            int l = i >> 6;            // 64 float4 per row
            int d = (i & 63) << 2;
            float m = mk[l];
            float4 o; o.x = v.x * m; o.y = v.y * m; o.z = v.z * m; o.w = v.w * m;
            *(float4*)&ctx[l * ROWP + d] = o;
        }
        // zero padded rows (incl. row pad)
        for (int i = tid; i < (LPAD - LSEQ) * ROWP; i += 256)
            ctx[LSEQ * ROWP + i] = 0.0f;
    }

    // ---- main GEMM: lin = ctxm @ (q*W)^T ; A[l] += sum_e relu(lin+b)*h[e] ----
    const int lrow    = lane & 15;
    const int halfsel = lane >> 4;       // 0: K {0,1}, 1: K {2,3}
    const int kphase  = halfsel << 1;

    for (int c = 0; c < 4; ++c) {
        __syncthreads();                 // wbuf safe to overwrite
        // stage W rows [c*64, c*64+64), pre-scaled by q (q-fold)
        {
            const float4* Wv = (const float4*)W;
            for (int i = tid; i < (WCHUNK * DDIM / 4); i += 256) {
                int rl = i >> 6;
                int d  = (i & 63) << 2;
                float4 v = Wv[(size_t)(c * WCHUNK + rl) * (DDIM / 4) + (i & 63)];
                const float4 qq = *(const float4*)&qv[d];
                v.x *= qq.x; v.y *= qq.y; v.z *= qq.z; v.w *= qq.w;
                *(float4*)&wbuf[rl * ROWP + d] = v;
            }
        }
        __syncthreads();

        // two M-stripes per wave: stripes (wave) and (wave+8); 13..15 discarded
        const int s0 = wave;
        const int s1 = wave + 8;
        const float* arow0 = ctx + ((s0 << 4) + lrow) * ROWP + kphase;
        const float* arow1 = ctx + ((s1 << 4) + lrow) * ROWP + kphase;  // in-bounds LDS
        const float* brow  = wbuf + lrow * ROWP + kphase;

        v8f acc0[4], acc1[4];
        #pragma unroll
        for (int t = 0; t < 4; ++t) {
            acc0[t] = (v8f){0,0,0,0,0,0,0,0};
            acc1[t] = (v8f){0,0,0,0,0,0,0,0};
        }

        #pragma unroll 4
        for (int k0 = 0; k0 < DDIM; k0 += 4) {
            v2f a0 = *(const v2f*)(arow0 + k0);
            v2f a1 = *(const v2f*)(arow1 + k0);
            v2f bf[4];
            #pragma unroll
            for (int t = 0; t < 4; ++t)
                bf[t] = *(const v2f*)(brow + t * 16 * ROWP + k0);
            #pragma unroll
            for (int t = 0; t < 4; ++t)
                acc0[t] = __builtin_amdgcn_wmma_f32_16x16x4_f32(
                    false, a0, false, bf[t], (short)0, acc0[t], false, false);
            #pragma unroll
            for (int t = 0; t < 4; ++t)
                acc1[t] = __builtin_amdgcn_wmma_f32_16x16x4_f32(
                    false, a1, false, bf[t], (short)0, acc1[t], false, false);
        }

        // epilogue: rs[r] = sum over 4 n-tiles of relu(lin + bias)*h, reduce, accumulate
        auto epilogue = [&](const v8f* acc, int m0, bool valid) {
            float rs[8];
            #pragma unroll
            for (int r = 0; r < 8; ++r) rs[r] = 0.0f;
            #pragma unroll
            for (int t = 0; t < 4; ++t) {
                int n = (c << 6) + (t << 4) + lrow;
                float bb = bv[n];
                float hh = hv[n];
                #pragma unroll
                for (int r = 0; r < 8; ++r) {
                    float lin = acc[t][r] + bb;
                    rs[r] += (lin > 0.0f ? lin : 0.0f) * hh;
                }
            }
            #pragma unroll
            for (int r = 0; r < 8; ++r) {
                rs[r] += __shfl_xor(rs[r], 1, 32);
                rs[r] += __shfl_xor(rs[r], 2, 32);
                rs[r] += __shfl_xor(rs[r], 4, 32);
                rs[r] += __shfl_xor(rs[r], 8, 32);
            }
            if (valid && lrow == 0) {
                int mrow = m0 + (halfsel << 3);
                #pragma unroll
                for (int r = 0; r < 8; ++r)
                    atomicAdd(&Abuf[mrow + r], rs[r]);   // ds_add_f32
            }
        };
        epilogue(acc0, s0 << 4, true);
        epilogue(acc1, s1 << 4, s1 < MTILES);
    }
    __syncthreads();

    // ---- beta-softmax with zero-row handling ----
    float a_l = 0.0f, e_l = 0.0f;
    if (tid < LSEQ) {
        a_l = Abuf[tid] * mk[tid];      // A = (act @ h) * mask
        e_l = __expf(a_l) * mk[tid];    // A_exp
    }
    float as = a_l, es = e_l;
    #pragma unroll
    for (int m = 1; m < 32; m <<= 1) {
        as += __shfl_xor(as, m, 32);
        es += __shfl_xor(es, m, 32);
    }
    if (lane == 0) { red[wave] = as; red[8 + wave] = es; }
    __syncthreads();
    float asum = 0.0f, denom = 0.0f;
    #pragma unroll
    for (int w = 0; w < 8; ++w) { asum += red[w]; denom += red[8 + w]; }
    float safe  = (denom > 0.0f) ? denom : 1.0f;
    float scale = 1.0f / sqrtf(safe);              // safe_denom ** (-BETA), BETA=0.5
    float attn_l = (asum != 0.0f) ? e_l * scale : 0.0f;
    if (tid < LPAD) attns[tid] = (tid < LSEQ) ? attn_l : 0.0f;
    if (tid < LSEQ)
        out[(size_t)BATCH * DDIM + (size_t)b * LSEQ + tid] = attn_l;
    __syncthreads();

    // ---- output[b,d] = sum_l attn[l] * ctx_masked[l,d] (ctx still in LDS) ----
    {
        float accd = 0.0f;
        const int d = tid;
        #pragma unroll 4
        for (int l = 0; l < LSEQ; ++l)
            accd += attns[l] * ctx[l * ROWP + d];
        out[(size_t)b * DDIM + d] = accd;
    }
}

extern "C" void kernel_launch(void* const* d_in, const int* in_sizes, int n_in,
                              void* d_out, int out_size, void* d_ws, size_t ws_size,
                              hipStream_t stream) {
    (void)in_sizes; (void)n_in; (void)out_size; (void)d_ws; (void)ws_size;
    const float* query   = (const float*)d_in[0];
    const float* context = (const float*)d_in[1];
    const float* cmask   = (const float*)d_in[2];
    const float* W       = (const float*)d_in[3];
    const float* bias    = (const float*)d_in[4];
    const float* hvec    = (const float*)d_in[5];
    float* out = (float*)d_out;

    (void)hipFuncSetAttribute((const void*)nais_attention_fused,
                              hipFuncAttributeMaxDynamicSharedMemorySize,
                              SMEM_BYTES);

    nais_attention_fused<<<dim3(BATCH), dim3(256), SMEM_BYTES, stream>>>(
        query, context, cmask, W, bias, hvec, out);
}